// FCOS_82626580840481
// MI455X (gfx1250) — compile-verified
//
#include <hip/hip_runtime.h>
#include <hip/hip_bf16.h>

#define NUM_CLASSES 80
#define NT_TOTAL    17064      // 12800+3200+800+208+56
#define NMS_BLOCK   1024       // 32 wave32s
#define PPT         17         // ceil(17064/1024)
#define POST_TOPK   100
#define NMS_THR     0.6f
#define CLASS_OFF   4096.0f

typedef unsigned int u32x4 __attribute__((ext_vector_type(4)));
typedef int          i32x8 __attribute__((ext_vector_type(8)));
typedef int          i32x4 __attribute__((ext_vector_type(4)));

// ---------------------------------------------------------------------------
// Stage 1: per-level decode. Batch on gridDim.y (no integer division);
// W is a power of two -> shift/mask for row/col. argmax over raw logits
// (sigmoid monotone, centerness constant per point) -> 1 sigmoid pair + sqrt.
// Class-offset shift folded into the stored box. Per-level sort is
// output-invariant -> skipped.
// ---------------------------------------------------------------------------
__global__ void __launch_bounds__(256) decode_level_kernel(
    const float* __restrict__ cls, const float* __restrict__ box,
    const float* __restrict__ ctr,
    float* __restrict__ sc_out, float4* __restrict__ ob_out,
    int* __restrict__ cl_out,
    int HW, int logW, float stride, int lvl_off)
{
  const int p = blockIdx.x * blockDim.x + threadIdx.x;   // point within level
  if (p >= HW) return;
  const int b = blockIdx.y;                              // image index

  const float* cbase = cls + (size_t)b * NUM_CLASSES * HW + p;
  float best = cbase[0];
  int   bi   = 0;
#pragma unroll 8
  for (int c = 1; c < NUM_CLASSES; ++c) {
    float v = cbase[(size_t)c * HW];
    if (v > best) { best = v; bi = c; }
  }
  float cv = ctr[(size_t)b * HW + p];
  // sqrt( sigmoid(best) * sigmoid(cv) )
  float s = __fsqrt_rn(__frcp_rn((1.0f + __expf(-best)) * (1.0f + __expf(-cv))));

  const float* bbase = box + (size_t)b * 4 * HW + p;
  float l = bbase[0]      * stride;
  float t = bbase[HW]     * stride;
  float r = bbase[2 * HW] * stride;
  float d = bbase[3 * HW] * stride;
  float px = (float)(p & ((1 << logW) - 1)) * stride + 0.5f * stride;
  float py = (float)(p >> logW)             * stride + 0.5f * stride;
  float x1 = fminf(fmaxf(px - l, 0.0f), 1023.0f);   // IMG_W-1
  float y1 = fminf(fmaxf(py - t, 0.0f),  799.0f);   // IMG_H-1
  float x2 = fminf(fmaxf(px + r, 0.0f), 1023.0f);
  float y2 = fminf(fmaxf(py + d, 0.0f),  799.0f);

  float off = (float)bi * CLASS_OFF;               // class-aware NMS shift
  int o = b * NT_TOTAL + lvl_off + p;
  sc_out[o] = s;
  cl_out[o] = bi;
  ob_out[o] = make_float4(x1 + off, y1 + off, x2 + off, y2 + off);
}

// ---------------------------------------------------------------------------
// Stage 2: sequential class-aware NMS, one workgroup per image.
// Scores DMA'd global->LDS via the Tensor Data Mover (wave 0 issues one
// tensor_load_to_lds, waits on TENSORcnt). Candidates live in registers
// (static indexing only -> no scratch). 100 serial rounds of block argmax
// (shfl_xor within wave32, LDS tree across 32 waves) + IoU suppression.
// ---------------------------------------------------------------------------
__global__ void __launch_bounds__(NMS_BLOCK) nms_kernel(
    const float* __restrict__ sc_g, const float4* __restrict__ ob_g,
    const int* __restrict__ cl_g,
    float* __restrict__ dets, float* __restrict__ cls_out)
{
  extern __shared__ float smem[];
  float* ssc  = smem;                          // NT_TOTAL scores (TDM target)
  float* rval = smem + NT_TOTAL;               // 32 per-wave maxima
  int*   ridx = (int*)(smem + NT_TOTAL + 32);  // 32 per-wave arg indices
  float* bwin = smem + NT_TOTAL + 64;          // winner shifted box (4 floats)

  const int b   = blockIdx.x;
  const int tid = threadIdx.x;
  const float NEG = -__builtin_huge_valf();

#if __has_builtin(__builtin_amdgcn_tensor_load_to_lds) && \
    __has_builtin(__builtin_amdgcn_s_wait_tensorcnt)
  if (tid < 32) {   // wave 0 only: TDM DMA of this image's score row (68 KB)
    unsigned lds_addr = (unsigned)(size_t)(void*)ssc;
    unsigned long long ga =
        (unsigned long long)(size_t)(sc_g + (size_t)b * NT_TOTAL);
    u32x4 g0;
    g0[0] = 1u;                                        // count=1, user mode
    g0[1] = lds_addr;                                  // lds_addr[31:0]
    g0[2] = (unsigned)(ga & 0xffffffffull);            // global_addr[31:0]
    g0[3] = (unsigned)((ga >> 32) & 0x01ffffffull)     // global_addr[56:32]
          | (2u << 30);                                // type = 2 ("image")
    i32x8 g1;
    g1[0] = (2 << 16);                                 // data_size = 4 bytes
    g1[1] = (int)((NT_TOTAL & 0xffff) << 16);          // tensor_dim0[15:0]
    g1[2] = (NT_TOTAL >> 16) | (1 << 16);              // dim0[31:16] | tensor_dim1=1
    g1[3] = (int)((NT_TOTAL & 0xffff) << 16);          // dim1[31:16]=0 | tile_dim0
    g1[4] = 1;                                         // tile_dim1=1, tile_dim2=0
    g1[5] = NT_TOTAL;                                  // tensor_dim0_stride[31:0]
    g1[6] = 0;
    g1[7] = 0;
    i32x4 gz4  = {0, 0, 0, 0};
    i32x8 gz8  = {0, 0, 0, 0, 0, 0, 0, 0};
    // clang-23 / therock-10.0 lane: 6-arg form (g0, g1, g2, g3, extra, cpol)
    __builtin_amdgcn_tensor_load_to_lds(g0, g1, gz4, gz4, gz8, 0);
    __builtin_amdgcn_s_wait_tensorcnt(0);
  }
#else
  for (int p = tid; p < NT_TOTAL; p += NMS_BLOCK)
    ssc[p] = sc_g[(size_t)b * NT_TOTAL + p];
#endif
  __syncthreads();

  // Per-thread candidate registers (strided ownership p = tid + k*1024).
  float bx1[PPT], by1[PPT], bx2[PPT], by2[PPT], msc[PPT], mco[PPT];
#pragma unroll
  for (int k = 0; k < PPT; ++k) {
    int p = tid + k * NMS_BLOCK;
    bx1[k] = by1[k] = bx2[k] = by2[k] = mco[k] = 0.0f;
    msc[k] = NEG;
    if (p < NT_TOTAL) {
      float4 bb = ob_g[(size_t)b * NT_TOTAL + p];     // coalesced b128 loads
      bx1[k] = bb.x; by1[k] = bb.y; bx2[k] = bb.z; by2[k] = bb.w;
      msc[k] = ssc[p];                                 // from LDS (TDM'd)
      mco[k] = (float)cl_g[(size_t)b * NT_TOTAL + p] * CLASS_OFF;
    }
  }

  for (int it = 0; it < POST_TOPK; ++it) {
    // ---- block argmax (value, lowest index on ties) ----
    float bv = msc[0];
    int   bp = tid;
#pragma unroll
    for (int k = 1; k < PPT; ++k) {
      int p = tid + k * NMS_BLOCK;
      if (msc[k] > bv) { bv = msc[k]; bp = p; }
    }
#pragma unroll
    for (int m = 16; m >= 1; m >>= 1) {
      float ov = __shfl_xor(bv, m, 32);
      int   op = __shfl_xor(bp, m, 32);
      if (ov > bv || (ov == bv && op < bp)) { bv = ov; bp = op; }
    }
    if ((tid & 31) == 0) { rval[tid >> 5] = bv; ridx[tid >> 5] = bp; }
    __syncthreads();
    if (tid < 32) {
      bv = rval[tid]; bp = ridx[tid];
#pragma unroll
      for (int m = 16; m >= 1; m >>= 1) {
        float ov = __shfl_xor(bv, m, 32);
        int   op = __shfl_xor(bp, m, 32);
        if (ov > bv || (ov == bv && op < bp)) { bv = ov; bp = op; }
      }
      if (tid == 0) { rval[0] = bv; ridx[0] = bp; }
    }
    __syncthreads();
    const float wv = rval[0];
    const int   wp = ridx[0];

    // ---- owning thread broadcasts winner box + writes detection ----
    if (tid == (wp & (NMS_BLOCK - 1))) {
      float sx1 = 0, sy1 = 0, sx2 = 0, sy2 = 0, sco = 0;
#pragma unroll
      for (int k = 0; k < PPT; ++k)                    // select, no dyn index
        if (tid + k * NMS_BLOCK == wp) {
          sx1 = bx1[k]; sy1 = by1[k]; sx2 = bx2[k]; sy2 = by2[k]; sco = mco[k];
        }
      bwin[0] = sx1; bwin[1] = sy1; bwin[2] = sx2; bwin[3] = sy2;
      bool valid = (wv >= 0.0f);                       // real scores >= 0
      float* dp = dets + ((size_t)b * POST_TOPK + it) * 5;
      dp[0] = valid ? sx1 - sco : 0.0f;
      dp[1] = valid ? sy1 - sco : 0.0f;
      dp[2] = valid ? sx2 - sco : 0.0f;
      dp[3] = valid ? sy2 - sco : 0.0f;
      dp[4] = valid ? wv : -1.0f;
      cls_out[b * POST_TOPK + it] = valid ? sco * (1.0f / CLASS_OFF) : -1.0f;
    }
    __syncthreads();

    // ---- suppression (register-local, shifted boxes => class-aware) ----
    const float wx1 = bwin[0], wy1 = bwin[1], wx2 = bwin[2], wy2 = bwin[3];
    const float wa  = (wx2 - wx1) * (wy2 - wy1);
#pragma unroll
    for (int k = 0; k < PPT; ++k) {
      int p = tid + k * NMS_BLOCK;
      float ix1 = fmaxf(wx1, bx1[k]), iy1 = fmaxf(wy1, by1[k]);
      float ix2 = fminf(wx2, bx2[k]), iy2 = fminf(wy2, by2[k]);
      float inter = fmaxf(ix2 - ix1, 0.0f) * fmaxf(iy2 - iy1, 0.0f);
      float area  = (bx2[k] - bx1[k]) * (by2[k] - by1[k]);
      float iou   = inter / (wa + area - inter + 1e-9f);
      if (iou > NMS_THR || p == wp) msc[k] = NEG;
    }
    __syncthreads();   // protect bwin/rval reuse next round
  }
}

// ---------------------------------------------------------------------------
extern "C" void kernel_launch(void* const* d_in, const int* in_sizes, int n_in,
                              void* d_out, int out_size, void* d_ws, size_t ws_size,
                              hipStream_t stream) {
  static const int   LH[5]  = {100, 50, 25, 13, 7};
  static const int   LW[5]  = {128, 64, 32, 16, 8};
  static const int   LGW[5] = {7, 6, 5, 4, 3};       // log2(LW)
  static const float LS[5]  = {8.f, 16.f, 32.f, 64.f, 128.f};

  // workspace: scores [16*N] f32 | shifted boxes [16*N] f32x4 | classes [16*N] i32
  float*  sc_ws = (float*)d_ws;
  float4* ob_ws = (float4*)((char*)d_ws + (size_t)16 * NT_TOTAL * sizeof(float));
  int*    cl_ws = (int*)((char*)ob_ws + (size_t)16 * NT_TOTAL * sizeof(float4));

  int off = 0;
  for (int l = 0; l < 5; ++l) {
    int HW = LH[l] * LW[l];
    hipLaunchKernelGGL(decode_level_kernel,
                       dim3((HW + 255) / 256, 16), dim3(256), 0, stream,
                       (const float*)d_in[l], (const float*)d_in[5 + l],
                       (const float*)d_in[10 + l],
                       sc_ws, ob_ws, cl_ws, HW, LGW[l], LS[l], off);
    off += HW;
  }

  float* dets    = (float*)d_out;                 // [16,100,5]
  float* cls_out = dets + 16 * POST_TOPK * 5;     // [16,100]
  size_t shmem = (size_t)(NT_TOTAL + 64 + 8) * sizeof(float);  // ~68.5 KB (WGP has 320 KB)
  (void)hipFuncSetAttribute((const void*)nms_kernel,
                            hipFuncAttributeMaxDynamicSharedMemorySize,
                            (int)shmem);
  hipLaunchKernelGGL(nms_kernel, dim3(16), dim3(NMS_BLOCK), shmem, stream,
                     sc_ws, ob_ws, cl_ws, dets, cls_out);
}